// SimpleMambaEncoderLayer_21071109554439
// MI455X (gfx1250) — compile-verified
//
#include <hip/hip_runtime.h>
#include <math.h>

typedef float v8f __attribute__((ext_vector_type(8)));
typedef float v2f __attribute__((ext_vector_type(2)));
typedef unsigned int v4u __attribute__((ext_vector_type(4)));
typedef int v8i __attribute__((ext_vector_type(8)));
typedef int v4i __attribute__((ext_vector_type(4)));

#define BATCHN  8
#define SIDE    32
#define SEQ     1024            // SIDE*SIDE
#define NROWS   (BATCHN*SEQ)    // 8192
#define DMODEL  96
#define DINNER  192
#define DSTATE  16
#define KDIRS   4
#define DTRANK  6
#define DFF     512
#define LNEPS   1e-5f
#define XDBL_LD 48              // padded (6 + 16 + 16 = 38 -> 48)

__device__ __forceinline__ float fast_silu(float v)
{
    // v * sigmoid(v) with v_rcp_f32 instead of the IEEE divide chain
    return v * __builtin_amdgcn_rcpf(1.f + __expf(-v));
}

// ---------------------------------------------------------------------------
// Zero-padding copy: dst[b,i,j] = (i<srcR && j<srcC) ? src[b,i,j] : 0
// ---------------------------------------------------------------------------
__global__ void pad2d(const float* __restrict__ src, float* __restrict__ dst,
                      int batch, int srcR, int srcC, int dstR, int dstC)
{
    int tid = blockIdx.x * blockDim.x + threadIdx.x;
    long total = (long)batch * dstR * dstC;
    if (tid >= total) return;
    int j = tid % dstC;
    int r = tid / dstC;
    int i = r % dstR;
    int b = r / dstR;
    dst[tid] = (i < srcR && j < srcC) ? src[((long)b * srcR + i) * srcC + j] : 0.f;
}

// ---------------------------------------------------------------------------
// Batched GEMM: Out[i,j] = epi( sum_k X[i,k]*W[j,k] + bias[j] ) + res[i,j]
//   - W row-major [N,K] with ld == K (contiguous N-panel)  -> TDM-staged in LDS
//   - block = 4 waves = 4 M-tiles sharing one N-panel of NB 16-wide tiles
//   - requires M%64==0, N%(16*NB)==0, K%KC==0, KC%4==0  (no guards anywhere)
// MODE: 0=none 1=silu 2=relu 3=softplus
// ---------------------------------------------------------------------------
template <int NB, int MODE, bool HASBIAS, bool HASRES>
__global__ void gemm_xwt(const float* __restrict__ X, long sX, int ldx,
                         const float* __restrict__ W, long sW, int wmod,
                         const float* __restrict__ bias, long sB, int bmod,
                         const float* __restrict__ resid, long sR, int ldr,
                         float* __restrict__ Out, long sO, int ldo,
                         int M, int N, int K, int KC)
{
    extern __shared__ float smem[];   // [16*NB rows][KC] weight panel chunk

    const int batch = blockIdx.y;
    const float* Xb = X + (long)batch * sX;
    const float* Wb = W + (long)(wmod ? (batch % wmod) : batch) * sW;
    float*       Ob = Out + (long)batch * sO;
    const float* biasb = HASBIAS ? (bias + (long)(bmod ? (batch % bmod) : batch) * sB) : nullptr;
    const float* resb  = HASRES  ? (resid + (long)batch * sR) : nullptr;

    const int groupsN = N / (16 * NB);
    const int bn = blockIdx.x % groupsN;
    const int bm = blockIdx.x / groupsN;
    const int j0 = bn * 16 * NB;
    const int warp = threadIdx.x >> 5;
    const int lane = threadIdx.x & 31;
    const int r16  = lane & 15;           // A: M row / B: N col
    const int half = lane >> 4;           // K sub-slot
    const int i0 = (bm * 4 + warp) << 4;
    const int nrows = 16 * NB;

    const float* xr = Xb + (long)(i0 + r16) * ldx;
    __builtin_prefetch(xr, 0, 3);

    v8f acc[NB];
#pragma unroll
    for (int nb = 0; nb < NB; ++nb) {
        v8f z = {0.f, 0.f, 0.f, 0.f, 0.f, 0.f, 0.f, 0.f};
        acc[nb] = z;
    }

    for (int kc0 = 0; kc0 < K; kc0 += KC) {
        // ---- TDM: wave 0 DMAs the [nrows x KC] weight panel chunk into LDS ----
        if (warp == 0) {
            const float* wpanel = Wb + (long)j0 * K + kc0;
            unsigned long long ga = (unsigned long long)(uintptr_t)wpanel;
            unsigned ldsa = (unsigned)(uintptr_t)smem;
            v4u g0;
            g0.x = 1u;                                        // count=1 (valid D#)
            g0.y = ldsa;                                      // lds_addr (bytes)
            g0.z = (unsigned)(ga & 0xFFFFFFFFull);            // global_addr[31:0]
            g0.w = (unsigned)((ga >> 32) & 0x1FFFFFFull)      // global_addr[56:32]
                 | (2u << 30);                                // type=2 ("image")
            v8i g1;
            g1[0] = 0x20000;                                  // data_size=2 -> 4B elems
            g1[1] = (int)((unsigned)KC << 16);                // tensor_dim0[15:0]
            g1[2] = (int)((unsigned)nrows << 16);             // dim0 hi | tensor_dim1[15:0]
            g1[3] = (int)((unsigned)KC << 16);                // dim1 hi | tile_dim0
            g1[4] = nrows;                                    // tile_dim1 | tile_dim2=0
            g1[5] = K;                                        // tensor_dim0_stride[31:0]
            g1[6] = 0;                                        // stride hi | dim1_stride lo
            g1[7] = 0;
            v4i gz4 = {0, 0, 0, 0};
            v8i gz8 = {0, 0, 0, 0, 0, 0, 0, 0};
            __builtin_amdgcn_tensor_load_to_lds(g0, g1, gz4, gz4, gz8, 0);
            __builtin_amdgcn_s_wait_tensorcnt(0);
        }
        __syncthreads();

        const int kend = kc0 + KC;
        for (int k0 = kc0; k0 < kend; k0 += 4) {
            const int ka = k0 + (half << 1);                  // even -> b64 loads
            v2f a;
            a.x = xr[ka];
            a.y = xr[ka + 1];
#pragma unroll
            for (int nb = 0; nb < NB; ++nb) {
                const float* bw = smem + (nb * 16 + r16) * KC + (ka - kc0);
                v2f b;
                b.x = bw[0];
                b.y = bw[1];
                acc[nb] = __builtin_amdgcn_wmma_f32_16x16x4_f32(
                    false, a, false, b, (short)0, acc[nb], false, false);
            }
        }
        __syncthreads();
    }

#pragma unroll
    for (int nb = 0; nb < NB; ++nb) {
#pragma unroll
        for (int r = 0; r < 8; ++r) {
            const int gi = i0 + r + (half << 3);   // D: VGPR r -> M=r (+8 hi half)
            const int gj = j0 + nb * 16 + r16;
            float v = acc[nb][r];
            if (HASBIAS) v += biasb[gj];
            if (MODE == 1)      v = fast_silu(v);                       // silu
            else if (MODE == 2) v = fmaxf(v, 0.f);                      // relu
            else if (MODE == 3) v = (v > 20.f) ? v : log1pf(__expf(v)); // softplus
            if (HASRES) v += resb[(long)gi * ldr + gj];
            Ob[(long)gi * ldo + gj] = v;
        }
    }
}

// ---------------------------------------------------------------------------
// Build xs_t[b,k,l,d] (4 scan orderings) from silu'd xx = xz[:, :192]
// ---------------------------------------------------------------------------
__global__ void build_xs(const float* __restrict__ xz, float* __restrict__ xs_t)
{
    const int total = BATCHN * KDIRS * SEQ * DINNER;
    int tid = blockIdx.x * blockDim.x + threadIdx.x;
    if (tid >= total) return;
    int d = tid % DINNER;
    int r = tid / DINNER;
    int l = r % SEQ; r /= SEQ;
    int k = r % KDIRS;
    int b = r / KDIRS;
    int lsrc;
    if (k == 0)      lsrc = l;
    else if (k == 1) lsrc = ((l & 31) << 5) | (l >> 5);             // (w,h) order
    else if (k == 2) lsrc = SEQ - 1 - l;                            // flip
    else { int lf = SEQ - 1 - l; lsrc = ((lf & 31) << 5) | (lf >> 5); }
    xs_t[tid] = xz[((long)(b * SEQ + lsrc)) * (2 * DINNER) + d];
}

// ---------------------------------------------------------------------------
// Selective scan: one wave = 2 channels (d, d+1); one lane per state (16 ea.)
// ---------------------------------------------------------------------------
__global__ void scan_kernel(const float* __restrict__ xs_t,    // [B*K, L, 192]
                            const float* __restrict__ delta_t, // [B*K, L, 192]
                            const float* __restrict__ xdbl,    // [B*K, L, 48]
                            const float* __restrict__ A_logs,  // [K, 192, 16]
                            const float* __restrict__ Dsv,     // [K, 192]
                            float* __restrict__ ys_t)          // [B*K, L, 192]
{
    const int nwaves = BATCHN * KDIRS * (DINNER / 2);  // 3072
    int wid  = (blockIdx.x * blockDim.x + threadIdx.x) >> 5;
    int lane = threadIdx.x & 31;
    if (wid >= nwaves) return;

    const int s  = lane & 15;       // state index
    const int ci = lane >> 4;       // which of the 2 channels
    const int d  = (wid % (DINNER / 2)) * 2 + ci;
    const int bk = wid / (DINNER / 2);          // b*4 + k
    const int k  = bk & 3;

    const float A  = -__expf(A_logs[((long)k * DINNER + d) * DSTATE + s]);
    const float Dd = Dsv[k * DINNER + d];
    const long rowbase = (long)bk * SEQ;

    float h = 0.f;
    for (int l = 0; l < SEQ; ++l) {
        const long rb = rowbase + l;
        const float dl = delta_t[rb * DINNER + d];
        const float u  = xs_t  [rb * DINNER + d];
        const float Bv = xdbl[rb * XDBL_LD + DTRANK + s];
        const float Cv = xdbl[rb * XDBL_LD + DTRANK + DSTATE + s];
        const float dA = __expf(dl * A);
        h = h * dA + dl * u * Bv;
        float y = h * Cv;
        y += __shfl_xor(y, 8, 16);
        y += __shfl_xor(y, 4, 16);
        y += __shfl_xor(y, 2, 16);
        y += __shfl_xor(y, 1, 16);
        if (s == 0) ys_t[rb * DINNER + d] = y + u * Dd;
    }
}

// ---------------------------------------------------------------------------
// Combine 4 scan directions + out_norm LayerNorm + multiply by z
// ---------------------------------------------------------------------------
__global__ void combine_ln_mul(const float* __restrict__ ys_t,
                               const float* __restrict__ xz,   // z = cols 192..383
                               const float* __restrict__ g,
                               const float* __restrict__ be,
                               float* __restrict__ yln)
{
    int wid  = (blockIdx.x * blockDim.x + threadIdx.x) >> 5;
    int lane = threadIdx.x & 31;
    if (wid >= NROWS) return;
    const int b = wid / SEQ, l = wid % SEQ;
    const int hh = l >> 5, ww = l & 31;
    const int lp = (ww << 5) | hh;
    const long base0 = ((long)(b * 4 + 0) * SEQ + l) * DINNER;
    const long base1 = ((long)(b * 4 + 1) * SEQ + lp) * DINNER;
    const long base2 = ((long)(b * 4 + 2) * SEQ + (SEQ - 1 - l)) * DINNER;
    const long base3 = ((long)(b * 4 + 3) * SEQ + (SEQ - 1 - lp)) * DINNER;

    float v[6];
    float s1 = 0.f, s2 = 0.f;
#pragma unroll
    for (int e = 0; e < 6; ++e) {
        const int d = e * 32 + lane;
        float t = ys_t[base0 + d] + ys_t[base1 + d] + ys_t[base2 + d] + ys_t[base3 + d];
        v[e] = t; s1 += t; s2 += t * t;
    }
    for (int off = 16; off; off >>= 1) {
        s1 += __shfl_xor(s1, off, 32);
        s2 += __shfl_xor(s2, off, 32);
    }
    const float mean = s1 * (1.f / DINNER);
    const float var  = s2 * (1.f / DINNER) - mean * mean;
    const float rstd = rsqrtf(var + LNEPS);
    const long zb = (long)wid * (2 * DINNER) + DINNER;
    const long ob = (long)wid * DINNER;
#pragma unroll
    for (int e = 0; e < 6; ++e) {
        const int d = e * 32 + lane;
        float t = (v[e] - mean) * rstd * g[d] + be[d];
        yln[ob + d] = t * xz[zb + d];
    }
}

// ---------------------------------------------------------------------------
// LayerNorm over rows of width 96 (one wave per row, 3 elems/lane)
// ---------------------------------------------------------------------------
__global__ void ln96(const float* __restrict__ in, float* __restrict__ out,
                     const float* __restrict__ g, const float* __restrict__ be)
{
    int wid  = (blockIdx.x * blockDim.x + threadIdx.x) >> 5;
    int lane = threadIdx.x & 31;
    if (wid >= NROWS) return;
    const long base = (long)wid * DMODEL;
    float v[3];
    float s1 = 0.f, s2 = 0.f;
#pragma unroll
    for (int e = 0; e < 3; ++e) {
        const int d = e * 32 + lane;
        float t = in[base + d];
        v[e] = t; s1 += t; s2 += t * t;
    }
    for (int off = 16; off; off >>= 1) {
        s1 += __shfl_xor(s1, off, 32);
        s2 += __shfl_xor(s2, off, 32);
    }
    const float mean = s1 * (1.f / DMODEL);
    const float var  = s2 * (1.f / DMODEL) - mean * mean;
    const float rstd = rsqrtf(var + LNEPS);
#pragma unroll
    for (int e = 0; e < 3; ++e) {
        const int d = e * 32 + lane;
        out[base + d] = (v[e] - mean) * rstd * g[d] + be[d];
    }
}

// ---------------------------------------------------------------------------
extern "C" void kernel_launch(void* const* d_in, const int* in_sizes, int n_in,
                              void* d_out, int out_size, void* d_ws, size_t ws_size,
                              hipStream_t stream)
{
    const float* x          = (const float*)d_in[0];   // [8,1024,96]
    const float* in_proj_w  = (const float*)d_in[1];   // [384,96]
    const float* x_proj_w   = (const float*)d_in[2];   // [4,38,192]
    const float* dt_projs_w = (const float*)d_in[3];   // [4,192,6]
    const float* dt_projs_b = (const float*)d_in[4];   // [4,192]
    const float* A_logs     = (const float*)d_in[5];   // [4,192,16]
    const float* Dsv        = (const float*)d_in[6];   // [4,192]
    const float* out_norm_g = (const float*)d_in[7];
    const float* out_norm_b = (const float*)d_in[8];
    const float* out_proj_w = (const float*)d_in[9];   // [96,192]
    const float* ln1_g      = (const float*)d_in[10];
    const float* ln1_b      = (const float*)d_in[11];
    const float* ln2_g      = (const float*)d_in[12];
    const float* ln2_b      = (const float*)d_in[13];
    const float* ff_w1      = (const float*)d_in[14];  // [512,96]
    const float* ff_b1      = (const float*)d_in[15];
    const float* ff_w2      = (const float*)d_in[16];  // [96,512]
    const float* ff_b2      = (const float*)d_in[17];
    float* out = (float*)d_out;

    // ---- workspace carve-up (floats) ----
    float* ws = (float*)d_ws;
    float* xz    = ws; ws += (long)NROWS * 384;                  // silu(in_proj)
    float* xs_t  = ws; ws += (long)BATCHN * KDIRS * SEQ * DINNER;
    float* xdbl  = ws; ws += (long)BATCHN * KDIRS * SEQ * XDBL_LD;
    float* delt  = ws; ws += (long)BATCHN * KDIRS * SEQ * DINNER;
    float* ys    = ws; ws += (long)BATCHN * KDIRS * SEQ * DINNER;
    float* yln   = ws; ws += (long)NROWS * DINNER;
    float* h1pre = ws; ws += (long)NROWS * DMODEL;
    float* h1    = ws; ws += (long)NROWS * DMODEL;
    float* ffa   = ws; ws += (long)NROWS * DFF;
    float* h2pre = ws; ws += (long)NROWS * DMODEL;
    float* xpw_p = ws; ws += (long)KDIRS * XDBL_LD * DINNER;     // padded x_proj_w
    float* dtw_p = ws; ws += (long)KDIRS * DINNER * 8;           // padded dt_projs_w

    const int TPB = 128;  // 4 waves

    // 0) zero-pad ragged weights: x_proj_w 38->48 rows, dt_projs_w K 6->8
    pad2d<<<(KDIRS * XDBL_LD * DINNER + 255) / 256, 256, 0, stream>>>(
        x_proj_w, xpw_p, KDIRS, 38, DINNER, XDBL_LD, DINNER);
    pad2d<<<(KDIRS * DINNER * 8 + 255) / 256, 256, 0, stream>>>(
        dt_projs_w, dtw_p, KDIRS, DINNER, DTRANK, DINNER, 8);

    // 1) xz = silu(x @ in_proj_w.T)       M=8192 N=384 K=96  NB=4 KC=96
    gemm_xwt<4, 1, false, false><<<dim3((NROWS/64) * (384/64), 1), TPB,
                                   64 * 96 * 4, stream>>>(
        x, 0, DMODEL, in_proj_w, 0, 0,
        nullptr, 0, 0, nullptr, 0, 0,
        xz, 0, 384, NROWS, 384, DMODEL, 96);

    // 2) gather 4 scan orderings
    build_xs<<<(BATCHN * KDIRS * SEQ * DINNER + 255) / 256, 256, 0, stream>>>(xz, xs_t);

    // 3) x_dbl = xs @ x_proj_w_pad[k].T   per (b,k): M=1024 N=48 K=192  NB=3 KC=192
    gemm_xwt<3, 0, false, false><<<dim3((SEQ/64) * (XDBL_LD/48), BATCHN * KDIRS), TPB,
                                   48 * 192 * 4, stream>>>(
        xs_t, (long)SEQ * DINNER, DINNER,
        xpw_p, (long)XDBL_LD * DINNER, KDIRS,
        nullptr, 0, 0, nullptr, 0, 0,
        xdbl, (long)SEQ * XDBL_LD, XDBL_LD, SEQ, XDBL_LD, DINNER, 192);

    // 4) delta = softplus(dts @ dtw_pad[k].T + dt_b[k])  M=1024 N=192 K=8  NB=4 KC=8
    //    (X = xdbl cols 0..7; cols 6,7 hit zero-padded weights -> exact)
    gemm_xwt<4, 3, true, false><<<dim3((SEQ/64) * (DINNER/64), BATCHN * KDIRS), TPB,
                                  64 * 8 * 4, stream>>>(
        xdbl, (long)SEQ * XDBL_LD, XDBL_LD,
        dtw_p, (long)DINNER * 8, KDIRS,
        dt_projs_b, DINNER, KDIRS, nullptr, 0, 0,
        delt, (long)SEQ * DINNER, DINNER, SEQ, DINNER, 8, 8);

    // 5) selective scan (sequential over L, parallel over 6144 channels)
    scan_kernel<<<(BATCHN * KDIRS * (DINNER/2)) / 4, TPB, 0, stream>>>(
        xs_t, delt, xdbl, A_logs, Dsv, ys);

    // 6) combine directions + out_norm LN + * z
    combine_ln_mul<<<NROWS / 4, TPB, 0, stream>>>(ys, xz, out_norm_g, out_norm_b, yln);

    // 7) h1pre = x + yln @ out_proj_w.T   M=8192 N=96 K=192  NB=3 KC=192
    gemm_xwt<3, 0, false, true><<<dim3((NROWS/64) * (DMODEL/48), 1), TPB,
                                  48 * 192 * 4, stream>>>(
        yln, 0, DINNER, out_proj_w, 0, 0,
        nullptr, 0, 0, x, 0, DMODEL,
        h1pre, 0, DMODEL, NROWS, DMODEL, DINNER, 192);

    // 8) h1 = ln1(h1pre)
    ln96<<<NROWS / 4, TPB, 0, stream>>>(h1pre, h1, ln1_g, ln1_b);

    // 9) ffa = relu(h1 @ ff_w1.T + b1)    M=8192 N=512 K=96  NB=4 KC=96
    gemm_xwt<4, 2, true, false><<<dim3((NROWS/64) * (DFF/64), 1), TPB,
                                  64 * 96 * 4, stream>>>(
        h1, 0, DMODEL, ff_w1, 0, 0,
        ff_b1, 0, 0, nullptr, 0, 0,
        ffa, 0, DFF, NROWS, DFF, DMODEL, 96);

    // 10) h2pre = h1 + (ffa @ ff_w2.T + b2)  M=8192 N=96 K=512  NB=3 KC=128
    gemm_xwt<3, 0, true, true><<<dim3((NROWS/64) * (DMODEL/48), 1), TPB,
                                 48 * 128 * 4, stream>>>(
        ffa, 0, DFF, ff_w2, 0, 0,
        ff_b2, 0, 0, h1, 0, DMODEL,
        h2pre, 0, DMODEL, NROWS, DMODEL, DFF, 128);

    // 11) out = ln2(h2pre)
    ln96<<<NROWS / 4, TPB, 0, stream>>>(h2pre, out, ln2_g, ln2_b);
}